// SparseBayesianLinear_63608465653841
// MI455X (gfx1250) — compile-verified
//
#include <hip/hip_runtime.h>
#include <hip/hip_bf16.h>

// ---------------------------------------------------------------------------
// SparseBayesianLinear forward, MI455X (gfx1250, wave32)
//   B=2, S=128, D_IN=1024, D_OUT=1024  ->  M = B*S = 256, N = 1024, K = 1024
//
// Outputs concatenated flat in d_out (float32):
//   [0 .. 262143]          y            [256,1024]
//   [262144 .. 524287]     sigma        [256,1024]  (== scores, same layout)
//   [524288 .. 268959743]  masked_w     [256,1024,1024]  (~1.07 GB -> dominates)
// ---------------------------------------------------------------------------

#define M_TOK   256
#define N_OUT   1024
#define K_IN    1024

typedef float v2f __attribute__((ext_vector_type(2)));
typedef float v4f __attribute__((ext_vector_type(4)));
typedef float v8f __attribute__((ext_vector_type(8)));

// ---------------------------------------------------------------------------
// 1) wp[i] = softplus(rho[i]) = logaddexp(rho, 0) = max(x,0) + log1p(exp(-|x|))
// ---------------------------------------------------------------------------
__global__ void sbl_softplus_kernel(const float* __restrict__ rho,
                                    float* __restrict__ wp, int n) {
    int i = blockIdx.x * blockDim.x + threadIdx.x;
    if (i < n) {
        float v = rho[i];
        wp[i] = fmaxf(v, 0.0f) + log1pf(expf(-fabsf(v)));
    }
}

// ---------------------------------------------------------------------------
// 2) thr[o] = mean(gate[o, :]) over K_IN
// ---------------------------------------------------------------------------
__global__ void sbl_rowmean_kernel(const float* __restrict__ gate,
                                   float* __restrict__ thr) {
    __shared__ float red[256];
    const int o   = blockIdx.x;
    const int tid = threadIdx.x;
    float s = 0.0f;
    const float* row = gate + (long long)o * K_IN;
    for (int d = tid; d < K_IN; d += 256) s += row[d];
    red[tid] = s;
    __syncthreads();
    for (int off = 128; off > 0; off >>= 1) {
        if (tid < off) red[tid] += red[tid + off];
        __syncthreads();
    }
    if (tid == 0) thr[o] = red[0] * (1.0f / (float)K_IN);
}

// ---------------------------------------------------------------------------
// 3) Dual GEMM via V_WMMA_F32_16X16X4_F32 (full fp32 — mask bit-exactness vs
//    the fp32 reference matters because scores feed a > threshold compare).
//    One wave computes one 16x16 tile of BOTH scores (x @ wp^T) and xw
//    (x @ mu^T), sharing the A-matrix fragment. 4 waves / block.
//
//    f32 WMMA VGPR layouts (ISA 7.12.2):
//      A 16x4 : lane L -> m = L&15, holds K = {2*(L>>4), 2*(L>>4)+1}  (float2)
//      B 4x16 : lane L -> n = L&15, holds K = {2*(L>>4), 2*(L>>4)+1}  (float2)
//               B[k][n] = W[n0+n][k]  (W row-major [O,D] -> contiguous float2)
//      C/D    : vgpr r, lane L -> m = r + 8*(L>>4), n = L&15
// ---------------------------------------------------------------------------
__global__ void sbl_dual_gemm_kernel(const float* __restrict__ x,     // [M,K]
                                     const float* __restrict__ wp,    // [N,K]
                                     const float* __restrict__ wmu,   // [N,K]
                                     const float* __restrict__ thr,   // [N]
                                     const float* __restrict__ bias,  // [N]
                                     float* __restrict__ out_y,       // [M,N]
                                     float* __restrict__ out_sigma,   // [M,N]
                                     unsigned char* __restrict__ mask)// [M,N]
{
    const int lane  = threadIdx.x & 31;
    const int wave  = threadIdx.x >> 5;            // 0..3
    const int mtile = blockIdx.x;                  // 16 tiles of M
    const int ntile = (blockIdx.y << 2) + wave;    // 64 tiles of N
    const int m0 = mtile << 4;
    const int n0 = ntile << 4;

    const int mn    = lane & 15;
    const int khalf = lane >> 4;                   // 0 or 1

    const float* xrow  = x   + (long long)(m0 + mn) * K_IN + 2 * khalf;
    const float* prow  = wp  + (long long)(n0 + mn) * K_IN + 2 * khalf;
    const float* murow = wmu + (long long)(n0 + mn) * K_IN + 2 * khalf;

    v8f accS = {};
    v8f accM = {};
    for (int k = 0; k < K_IN; k += 4) {
        v2f a  = *(const v2f*)(xrow  + k);
        v2f bp = *(const v2f*)(prow  + k);
        v2f bm = *(const v2f*)(murow + k);
        accS = __builtin_amdgcn_wmma_f32_16x16x4_f32(
                   false, a, false, bp, (short)0, accS, false, false);
        accM = __builtin_amdgcn_wmma_f32_16x16x4_f32(
                   false, a, false, bm, (short)0, accM, false, false);
    }

    const int   col  = n0 + mn;
    const float tcol = thr[col];
    const float bcol = bias[col];
#pragma unroll
    for (int r = 0; r < 8; ++r) {
        const int row = m0 + r + 8 * khalf;
        const float s  = accS[r];
        const float xw = accM[r];
        const bool  mk = s > tcol;
        const long long idx = (long long)row * N_OUT + col;
        out_sigma[idx] = s;
        out_y[idx]     = (mk ? xw : 0.0f) + bcol;
        mask[idx]      = mk ? (unsigned char)1 : (unsigned char)0;
    }
}

// ---------------------------------------------------------------------------
// 4) masked_w[row, :] = mask[row] ? mu[row % N_OUT, :] : 0     (row = t*N + o)
//    ~1.07 GB of writes: the whole-problem roofline. mu (4 MB) stays hot in
//    L2 (192 MB); the output exceeds L2, so use non-temporal B128 stores.
//    256 threads/block, ROWS_PER_BLOCK rows, 1 float4 per thread per row.
// ---------------------------------------------------------------------------
#define ROWS_PER_BLOCK 16

__global__ void sbl_expand_kernel(const float* __restrict__ mu,        // [N,K]
                                  const unsigned char* __restrict__ mask,
                                  float* __restrict__ out_mw) {
    const int tid = threadIdx.x;   // 0..255 -> one float4 (16B) per row
    const long long rowBase = (long long)blockIdx.x * ROWS_PER_BLOCK;
#pragma unroll
    for (int r = 0; r < ROWS_PER_BLOCK; ++r) {
        const long long row = rowBase + r;
        const int o = (int)(row & (N_OUT - 1));
        v4f v = {};
        if (mask[row]) {
            v = ((const v4f*)(mu + (long long)o * K_IN))[tid];
        }
        v4f* dst = (v4f*)(out_mw + row * K_IN) + tid;
        __builtin_nontemporal_store(v, dst);
    }
}

// ---------------------------------------------------------------------------
// launch
// ---------------------------------------------------------------------------
extern "C" void kernel_launch(void* const* d_in, const int* in_sizes, int n_in,
                              void* d_out, int out_size, void* d_ws, size_t ws_size,
                              hipStream_t stream) {
    const float* x    = (const float*)d_in[0];  // [2,128,1024]
    const float* wmu  = (const float*)d_in[1];  // [1024,1024]
    const float* rho  = (const float*)d_in[2];  // [1024,1024]
    const float* gate = (const float*)d_in[3];  // [1024,1024]
    const float* bias = (const float*)d_in[4];  // [1024]

    float* out   = (float*)d_out;
    float* out_y     = out;
    float* out_sigma = out + (long long)M_TOK * N_OUT;
    float* out_mw    = out + 2LL * M_TOK * N_OUT;

    // workspace layout
    char* ws = (char*)d_ws;
    float*         wp   = (float*)ws;                                  // 4 MB
    float*         thr  = (float*)(ws + sizeof(float) * N_OUT * K_IN); // 4 KB
    unsigned char* mask = (unsigned char*)(ws + sizeof(float) * N_OUT * K_IN
                                              + sizeof(float) * N_OUT); // 256 KB

    // 1) softplus(rho) -> wp
    {
        const int n = N_OUT * K_IN;
        sbl_softplus_kernel<<<(n + 255) / 256, 256, 0, stream>>>(rho, wp, n);
    }
    // 2) thresholds
    sbl_rowmean_kernel<<<N_OUT, 256, 0, stream>>>(gate, thr);
    // 3) dual WMMA GEMM + gate epilogue
    {
        dim3 grid(M_TOK / 16, N_OUT / 64);   // (16, 16); 4 waves/block, 1 tile/wave
        sbl_dual_gemm_kernel<<<grid, 128, 0, stream>>>(
            x, wp, wmu, thr, bias, out_y, out_sigma, mask);
    }
    // 4) 1.07 GB masked-weight expansion (bandwidth bound)
    {
        const long long nrows = (long long)M_TOK * N_OUT;   // 262144
        sbl_expand_kernel<<<(int)(nrows / ROWS_PER_BLOCK), 256, 0, stream>>>(
            wmu, mask, out_mw);
    }
}